// NGCN_71159018160549
// MI455X (gfx1250) — compile-verified
//
#include <hip/hip_runtime.h>
#include <hip/hip_bf16.h>

typedef __attribute__((ext_vector_type(2))) float v2f;
typedef __attribute__((ext_vector_type(8))) float v8f;

#define DIN   100
#define HDIM  16
#define CDIM  40
#define NCELL 12
#define FLATK (NCELL * CDIM)   /* 480 */
#define BN_EPS 1e-5f

static __device__ __forceinline__ v8f wmma_f32(v2f a, v2f b, v8f c) {
  // D = A(16x4 f32) * B(4x16 f32) + C(16x16 f32)
  return __builtin_amdgcn_wmma_f32_16x16x4_f32(false, a, false, b, (short)0, c,
                                               false, false);
}

// ---------------------------------------------------------------- degree / dinv
__global__ void k_deg_init(float* __restrict__ deg, int n) {
  int i = blockIdx.x * blockDim.x + threadIdx.x;
  if (i < n) deg[i] = 1.0f;                       // self loop contributes 1
}

__global__ void k_deg_accum(const int* __restrict__ er, float* __restrict__ deg,
                            int nE) {
  int i = blockIdx.x * blockDim.x + threadIdx.x;
  if (i < nE) atomicAdd(&deg[er[i]], 1.0f);
}

__global__ void k_dinv(float* __restrict__ deg, int n) {
  int i = blockIdx.x * blockDim.x + threadIdx.x;
  if (i < n) deg[i] = rsqrtf(deg[i]);             // deg >= 1 always
}

// ---------------------------------------------------------------- row L2 norm
__global__ __launch_bounds__(128) void k_l2norm(const float* __restrict__ x,
                                                float* __restrict__ xn, int n) {
  int node = blockIdx.x * 4 + (threadIdx.x >> 5);
  int lane = threadIdx.x & 31;
  if (node >= n) return;
  const float* xr = x + (size_t)node * DIN;
  float v0 = xr[lane];
  float v1 = xr[lane + 32];
  float v2 = xr[lane + 64];
  float v3 = (lane < DIN - 96) ? xr[lane + 96] : 0.0f;
  float s = v0 * v0 + v1 * v1 + v2 * v2 + v3 * v3;
#pragma unroll
  for (int off = 16; off > 0; off >>= 1) s += __shfl_xor(s, off, 32);
  float inv = 1.0f / fmaxf(sqrtf(s), 1e-12f);
  float* xo = xn + (size_t)node * DIN;
  xo[lane]      = v0 * inv;
  xo[lane + 32] = v1 * inv;
  xo[lane + 64] = v2 * inv;
  if (lane < DIN - 96) xo[lane + 96] = v3 * inv;
}

// ---------------------------------------------------------------- SpMM (A_hat)
// out = dinv^2 * in   (self-loop term; also initializes the output buffer)
template <int F>
__global__ void k_spmm_self(const float* __restrict__ in, float* __restrict__ out,
                            const float* __restrict__ dinv, int n) {
  int idx = blockIdx.x * blockDim.x + threadIdx.x;
  if (idx >= n * F) return;
  int r = idx / F;
  float d = dinv[r];
  out[idx] = d * d * in[idx];
}

// F=100: one wave per edge; row/col/weight loaded once, lanes stream features
__global__ __launch_bounds__(256) void k_spmm_scatter100(
    const int* __restrict__ er, const int* __restrict__ ec,
    const float* __restrict__ dinv, const float* __restrict__ in,
    float* __restrict__ out, int nE) {
  int wid  = (blockIdx.x * blockDim.x + threadIdx.x) >> 5;
  int lane = threadIdx.x & 31;
  if (wid >= nE) return;
  int r = er[wid], c = ec[wid];
  float v = dinv[r] * dinv[c];
  const float* ip = in + (size_t)c * DIN;
  float* op = out + (size_t)r * DIN;
  atomicAdd(op + lane,      v * ip[lane]);
  atomicAdd(op + lane + 32, v * ip[lane + 32]);
  atomicAdd(op + lane + 64, v * ip[lane + 64]);
  if (lane < DIN - 96) atomicAdd(op + lane + 96, v * ip[lane + 96]);
}

// F=16: thread per (edge,feature); half-wave per edge keeps stores coalesced
__global__ void k_spmm_scatter16(const int* __restrict__ er,
                                 const int* __restrict__ ec,
                                 const float* __restrict__ dinv,
                                 const float* __restrict__ in,
                                 float* __restrict__ out, int nE) {
  int t = blockIdx.x * blockDim.x + threadIdx.x;
  int e = t >> 4, f = t & 15;
  if (e >= nE) return;
  int r = er[e], c = ec[e];
  float v = dinv[r] * dinv[c];
  atomicAdd(out + (size_t)r * HDIM + f, v * in[(size_t)c * HDIM + f]);
}

// ---------------------------------------------------------------- GEMM1 (x3 replicas)
// hid_r = ReLU(BN_r(xs @ W1_r + b1_r)) for r=0..2 sharing A fragments.
// Param pointers are pre-offset to cell index h; replica stride = 4 cells.
__device__ __forceinline__ void epi1(v8f acc, const float* __restrict__ b1,
                                     const float* __restrict__ g,
                                     const float* __restrict__ be,
                                     const float* __restrict__ mu,
                                     const float* __restrict__ va,
                                     float* __restrict__ hid, int m, int rbase) {
  float sc = g[m] * rsqrtf(va[m] + BN_EPS);
  float sh = (b1[m] - mu[m]) * sc + be[m];
#pragma unroll
  for (int j = 0; j < 8; j++)
    hid[(size_t)(rbase + j) * HDIM + m] = fmaxf(acc[j] * sc + sh, 0.0f);
}

__global__ __launch_bounds__(128) void k_gemm1x3(
    const float* __restrict__ X, const float* __restrict__ W1,
    const float* __restrict__ b1, const float* __restrict__ gamma,
    const float* __restrict__ beta, const float* __restrict__ mean,
    const float* __restrict__ var, float* __restrict__ hid0,
    float* __restrict__ hid1, float* __restrict__ hid2, int nTiles) {
  __shared__ float sW[3 * DIN * HDIM];      // 19.2 KB: W1 panels, 3 replicas
  for (int t = threadIdx.x; t < 3 * DIN * HDIM; t += blockDim.x) {
    int rr = t / (DIN * HDIM), off = t - rr * (DIN * HDIM);
    sW[t] = W1[(size_t)rr * 4 * DIN * HDIM + off];
  }
  __syncthreads();

  int lane = threadIdx.x & 31;
  int tile = blockIdx.x * 4 + (threadIdx.x >> 5);
  if (tile >= nTiles) return;

  int m  = lane & 15;                       // A row / B col / C col
  int kk = (lane >> 4) << 1;                // 0 or 2
  const float* xrow = X + (size_t)(tile * 16 + m) * DIN + kk;

  v8f a0 = {}, a1 = {}, a2 = {};
#pragma unroll 5
  for (int k = 0; k < DIN; k += 4) {
    v2f a; a.x = xrow[k]; a.y = xrow[k + 1];
    int i0 = (k + kk) * HDIM + m, i1 = (k + kk + 1) * HDIM + m;
    v2f b;
    b.x = sW[i0];                      b.y = sW[i1];                      a0 = wmma_f32(a, b, a0);
    b.x = sW[DIN * HDIM + i0];         b.y = sW[DIN * HDIM + i1];         a1 = wmma_f32(a, b, a1);
    b.x = sW[2 * DIN * HDIM + i0];     b.y = sW[2 * DIN * HDIM + i1];     a2 = wmma_f32(a, b, a2);
  }

  const int PS = 4 * HDIM;                  // replica param stride (cells of 16)
  int rbase = tile * 16 + ((lane >> 4) << 3);
  epi1(a0, b1,          gamma,          beta,          mean,          var,          hid0, m, rbase);
  epi1(a1, b1 + PS,     gamma + PS,     beta + PS,     mean + PS,     var + PS,     hid1, m, rbase);
  epi1(a2, b1 + 2 * PS, gamma + 2 * PS, beta + 2 * PS, mean + 2 * PS, var + 2 * PS, hid2, m, rbase);
}

// ---------------------------------------------------------------- GEMM2
// cell_out = hid @ W2 + b2;  [N,16]x[16,40] -> d_out cell region [N,12,40]
__global__ __launch_bounds__(128) void k_gemm2(const float* __restrict__ hid,
                                               const float* __restrict__ W2,
                                               const float* __restrict__ b2,
                                               float* __restrict__ cellout,
                                               int cell, int nTiles) {
  __shared__ float sW[HDIM * 48];           // 40 cols padded to 48
  for (int t = threadIdx.x; t < HDIM * 48; t += blockDim.x) {
    int k = t / 48, n = t - k * 48;
    sW[t] = (n < CDIM) ? W2[k * CDIM + n] : 0.0f;
  }
  __syncthreads();

  int lane = threadIdx.x & 31;
  int tile = blockIdx.x * 4 + (threadIdx.x >> 5);
  if (tile >= nTiles) return;

  int m  = lane & 15;
  int kk = (lane >> 4) << 1;
  const float* hrow = hid + (size_t)(tile * 16 + m) * HDIM + kk;
  v2f a[4];
#pragma unroll
  for (int ks = 0; ks < 4; ks++) { a[ks].x = hrow[ks * 4]; a[ks].y = hrow[ks * 4 + 1]; }

  int rbase = tile * 16 + ((lane >> 4) << 3);
#pragma unroll
  for (int nt = 0; nt < 3; nt++) {
    v8f acc = {};
#pragma unroll
    for (int ks = 0; ks < 4; ks++) {
      int k = ks * 4 + kk;
      v2f b; b.x = sW[k * 48 + nt * 16 + m]; b.y = sW[(k + 1) * 48 + nt * 16 + m];
      acc = wmma_f32(a[ks], b, acc);
    }
    int n = nt * 16 + m;
    if (n < CDIM) {
      float bb = b2[n];
#pragma unroll
      for (int j = 0; j < 8; j++)
        cellout[((size_t)(rbase + j) * NCELL + cell) * CDIM + n] = acc[j] + bb;
    }
  }
}

// ---------------------------------------------------------------- GEMM3
// output = ReLU(cell_outputs.reshape(N,480)) @ Wout + bout.
// Single pass over A: all 3 n-tiles per wave; full padded Wout panel in LDS
// (92 KB -- CDNA5 WGP has 320 KB LDS, ~3 blocks/WGP resident).
__global__ __launch_bounds__(128) void k_gemm3(const float* __restrict__ flat,
                                               const float* __restrict__ Wout,
                                               const float* __restrict__ bout,
                                               float* __restrict__ out,
                                               int nTiles) {
  __shared__ float sW[FLATK * 48];          // 92160 B
  for (int t = threadIdx.x; t < FLATK * 48; t += blockDim.x) {
    int k = t / 48, n = t - k * 48;
    sW[t] = (n < CDIM) ? Wout[k * CDIM + n] : 0.0f;
  }
  __syncthreads();

  int lane = threadIdx.x & 31;
  int tile = blockIdx.x * 4 + (threadIdx.x >> 5);
  if (tile >= nTiles) return;

  int m  = lane & 15;
  int kk = (lane >> 4) << 1;
  const float* arow = flat + (size_t)(tile * 16 + m) * FLATK + kk;

  v8f c0 = {}, c1 = {}, c2 = {};
#pragma unroll 4
  for (int k = 0; k < FLATK; k += 4) {
    v2f a; a.x = fmaxf(arow[k], 0.0f); a.y = fmaxf(arow[k + 1], 0.0f);
    int i0 = (k + kk) * 48 + m, i1 = (k + kk + 1) * 48 + m;
    v2f b;
    b.x = sW[i0];      b.y = sW[i1];      c0 = wmma_f32(a, b, c0);
    b.x = sW[i0 + 16]; b.y = sW[i1 + 16]; c1 = wmma_f32(a, b, c1);
    b.x = sW[i0 + 32]; b.y = sW[i1 + 32]; c2 = wmma_f32(a, b, c2);
  }

  int rbase = tile * 16 + ((lane >> 4) << 3);
  v8f accs[3] = {c0, c1, c2};
#pragma unroll
  for (int nt = 0; nt < 3; nt++) {
    int n = nt * 16 + m;
    if (n < CDIM) {
      float bb = bout[n];
#pragma unroll
      for (int j = 0; j < 8; j++)
        out[(size_t)(rbase + j) * CDIM + n] = accs[nt][j] + bb;
    }
  }
}

// ================================================================ launcher
extern "C" void kernel_launch(void* const* d_in, const int* in_sizes, int n_in,
                              void* d_out, int out_size, void* d_ws, size_t ws_size,
                              hipStream_t stream) {
  const float* x     = (const float*)d_in[0];
  const int*   er    = (const int*)d_in[1];
  const int*   ec    = (const int*)d_in[2];
  const float* W1    = (const float*)d_in[3];   // [12,100,16]
  const float* b1    = (const float*)d_in[4];   // [12,16]
  const float* gamma = (const float*)d_in[5];
  const float* beta  = (const float*)d_in[6];
  const float* bmean = (const float*)d_in[7];
  const float* bvar  = (const float*)d_in[8];
  const float* W2    = (const float*)d_in[9];   // [12,16,40]
  const float* b2    = (const float*)d_in[10];  // [12,40]
  const float* Wout  = (const float*)d_in[11];  // [480,40]
  const float* bout  = (const float*)d_in[12];  // [40]

  const int nN = in_sizes[0] / DIN;             // 50000
  const int nE = in_sizes[1];                   // 800000
  const int nTiles = nN / 16;                   // 3125 (exact)

  // workspace layout (floats)
  float* f = (float*)d_ws;
  size_t o = 0;
  float* dinv = f + o; o += (size_t)((nN + 63) & ~63);
  float* xsA  = f + o; o += (size_t)nN * DIN;
  float* xsB  = f + o; o += (size_t)nN * DIN;
  float* hidR[3];
  for (int r = 0; r < 3; r++) { hidR[r] = f + o; o += (size_t)nN * HDIM; }
  float* hidS = f + o; o += (size_t)nN * HDIM;

  float* outp     = (float*)d_out;                      // [N,40]
  float* cellbase = outp + (size_t)nN * CDIM;           // [N,12,40]

  const int B = 256;
  // --- normalized adjacency weights
  k_deg_init <<<(nN + B - 1) / B, B, 0, stream>>>(dinv, nN);
  k_deg_accum<<<(nE + B - 1) / B, B, 0, stream>>>(er, dinv, nE);
  k_dinv     <<<(nN + B - 1) / B, B, 0, stream>>>(dinv, nN);

  // --- row-L2 normalize input features
  k_l2norm<<<(nN + 3) / 4, 128, 0, stream>>>(x, xsA, nN);

  const int gemmGrid = (nTiles + 3) / 4;
  float* cur = xsA; float* nxt = xsB;

  for (int h = 0; h <= 3; h++) {
    // all 3 replicas of this hop share A = cur
    k_gemm1x3<<<gemmGrid, 128, 0, stream>>>(
        cur, W1 + (size_t)h * DIN * HDIM, b1 + h * HDIM, gamma + h * HDIM,
        beta + h * HDIM, bmean + h * HDIM, bvar + h * HDIM,
        hidR[0], hidR[1], hidR[2], nTiles);

    for (int r = 0; r < 3; r++) {
      int i = r * 4 + h;
      float* hc = hidR[r]; float* hn = hidS;
      for (int s = 0; s < h; s++) {
        k_spmm_self<HDIM><<<((size_t)nN * HDIM + B - 1) / B, B, 0, stream>>>(
            hc, hn, dinv, nN);
        k_spmm_scatter16<<<((size_t)nE * HDIM + B - 1) / B, B, 0, stream>>>(
            er, ec, dinv, hc, hn, nE);
        float* t = hc; hc = hn; hn = t;
      }
      k_gemm2<<<gemmGrid, 128, 0, stream>>>(hc, W2 + (size_t)i * HDIM * CDIM,
                                            b2 + i * CDIM, cellbase, i, nTiles);
    }
    if (h < 3) {
      k_spmm_self<DIN><<<((size_t)nN * DIN + B - 1) / B, B, 0, stream>>>(
          cur, nxt, dinv, nN);
      k_spmm_scatter100<<<((size_t)nE * 32 + B - 1) / B, B, 0, stream>>>(
          er, ec, dinv, cur, nxt, nE);
      float* t = cur; cur = nxt; nxt = t;
    }
  }

  // --- final fusion layer (single pass over flat activations)
  k_gemm3<<<gemmGrid, 128, 0, stream>>>(cellbase, Wout, bout, outp, nTiles);
}